// PillarEncoder_64922725646716
// MI455X (gfx1250) — compile-verified
//
#include <hip/hip_runtime.h>

typedef __attribute__((ext_vector_type(2))) float v2f;
typedef __attribute__((ext_vector_type(8))) float v8f;

#define VXc   0.16f
#define VYc   0.16f
#define XOFFc 0.08f
#define YOFFc (0.08f - 39.68f)
#define XLc   432
#define YLc   496
#define NPTSc 32
#define COUTc 64
#define EPSc  1e-3f
#define NTOTc (40000.0f * 32.0f)

// Build the 9 per-point features (padded to 12) for one pillar.
// lane = point index within the pillar (wave32: one wave == one pillar).
__device__ __forceinline__ void pillar_features(
    const float4 pt, int lane, int n, float cx, float cy, float f[12])
{
    // cluster mean: reference sums ALL 32 points, divides by npoints
    float sx = pt.x, sy = pt.y, sz = pt.z;
    #pragma unroll
    for (int off = 16; off > 0; off >>= 1) {
        sx += __shfl_xor(sx, off, 32);
        sy += __shfl_xor(sy, off, 32);
        sz += __shfl_xor(sz, off, 32);
    }
    float invn = 1.0f / (float)n;
    float mx = sx * invn, my = sy * invn, mz = sz * invn;
    float msk = (lane < n) ? 1.0f : 0.0f;
    f[0] = pt.x * msk;
    f[1] = pt.y * msk;
    f[2] = pt.z * msk;
    f[3] = pt.w * msk;
    f[4] = (pt.x - mx) * msk;
    f[5] = (pt.y - my) * msk;
    f[6] = (pt.z - mz) * msk;
    f[7] = (pt.x - cx) * msk;
    f[8] = (pt.y - cy) * msk;
    f[9] = 0.f; f[10] = 0.f; f[11] = 0.f;
}

// Pass 1: accumulate Sum(f) [9] and Sum(f_i * f_j), i<=j [45] -> gstats[0..53]
__global__ void __launch_bounds__(256) pillar_stats_kernel(
    const float4* __restrict__ pts, const int* __restrict__ coors,
    const int* __restrict__ npts, float* __restrict__ gstats)
{
    __shared__ float sm[54];
    int tid = threadIdx.x;
    if (tid < 54) sm[tid] = 0.f;
    __syncthreads();

    int w = tid >> 5, lane = tid & 31;
    int p = blockIdx.x * 8 + w;
    float4 pt = pts[(size_t)p * NPTSc + lane];
    int n = npts[p];
    float cx = (float)coors[p * 4 + 1] * VXc + XOFFc;
    float cy = (float)coors[p * 4 + 2] * VYc + YOFFc;
    float f[12];
    pillar_features(pt, lane, n, cx, cy, f);

    if (lane < n) {  // masked points are exact zeros: skip
        #pragma unroll
        for (int i = 0; i < 9; ++i) atomicAdd(&sm[i], f[i]);
        int cnt = 9;
        #pragma unroll
        for (int i = 0; i < 9; ++i)
            #pragma unroll
            for (int j = i; j < 9; ++j)
                atomicAdd(&sm[cnt++], f[i] * f[j]);
    }
    __syncthreads();
    if (tid < 54) atomicAdd(&gstats[tid], sm[tid]);
}

// Pass 2: fold moments + gamma/beta into per-channel y = a*x + b
__global__ void __launch_bounds__(64) pillar_finalize_kernel(
    const float* __restrict__ gstats, const float* __restrict__ W,
    const float* __restrict__ gamma, const float* __restrict__ beta,
    float* __restrict__ ab)
{
    int o = threadIdx.x;
    float w[9];
    #pragma unroll
    for (int c = 0; c < 9; ++c) w[c] = W[o * 9 + c];

    const float invN = 1.0f / NTOTc;
    float mean = 0.f;
    #pragma unroll
    for (int c = 0; c < 9; ++c) mean += w[c] * gstats[c];
    mean *= invN;

    float ex2 = 0.f;
    int cnt = 9;
    #pragma unroll
    for (int i = 0; i < 9; ++i)
        #pragma unroll
        for (int j = i; j < 9; ++j) {
            float m = gstats[cnt++];
            ex2 += ((i == j) ? 1.0f : 2.0f) * w[i] * w[j] * m;
        }
    ex2 *= invN;

    float var = ex2 - mean * mean;
    float a = gamma[o] * rsqrtf(var + EPSc);
    float b = beta[o] - mean * a;
    ab[o] = a;
    ab[64 + o] = b;
}

// Pass 3: per-pillar WMMA GEMM (32x12 @ 12x64, fp32 K=4 tiles) fused with
// affine + ReLU + max-pool over 32 points + scatter to canvas (B,64,Y,X).
__global__ void __launch_bounds__(128) pillar_main_kernel(
    const float4* __restrict__ pts, const int* __restrict__ coors,
    const int* __restrict__ npts, const float* __restrict__ W,
    const float* __restrict__ ab, float* __restrict__ out)
{
    __shared__ float sf[4][NPTSc][12];   // stride 12 -> conflict-free reads
    int tid = threadIdx.x;
    int w = tid >> 5, lane = tid & 31;
    int p = blockIdx.x * 4 + w;

    float4 pt = pts[(size_t)p * NPTSc + lane];
    int n  = npts[p];
    int bb = coors[p * 4 + 0];
    int xi = coors[p * 4 + 1];
    int yi = coors[p * 4 + 2];
    float cx = (float)xi * VXc + XOFFc;
    float cy = (float)yi * VYc + YOFFc;

    float f[12];
    pillar_features(pt, lane, n, cx, cy, f);
    #pragma unroll
    for (int k = 0; k < 12; ++k) sf[w][lane][k] = f[k];

    // B operand: B[k][n] = W[n][k] (zero-padded k>=9).
    // Lane layout: lane = column (N = nt*16 + lane&15); lanes 16-31 hold the
    // upper K-pair of each K=4 step (mirrors 16x4 fp32 A layout).
    int col   = lane & 15;
    int khalf = (lane >> 4) << 1;
    v2f Bop[3][4];
    float aco[4], bco[4];
    #pragma unroll
    for (int nt = 0; nt < 4; ++nt) {
        int ch = nt * 16 + col;
        aco[nt] = ab[ch];
        bco[nt] = ab[64 + ch];
        #pragma unroll
        for (int kk = 0; kk < 3; ++kk) {
            int k0 = kk * 4 + khalf;
            v2f b;
            b.x = (k0     < 9) ? W[ch * 9 + k0]     : 0.f;
            b.y = (k0 + 1 < 9) ? W[ch * 9 + k0 + 1] : 0.f;
            Bop[kk][nt] = b;
        }
    }
    __syncthreads();

    v8f acc[2][4];
    #pragma unroll
    for (int mt = 0; mt < 2; ++mt)
        #pragma unroll
        for (int nt = 0; nt < 4; ++nt)
            acc[mt][nt] = (v8f){0.f, 0.f, 0.f, 0.f, 0.f, 0.f, 0.f, 0.f};

    // 24 x v_wmma_f32_16x16x4_f32, straight-line, full EXEC
    #pragma unroll
    for (int kk = 0; kk < 3; ++kk) {
        int k0 = kk * 4 + khalf;
        #pragma unroll
        for (int mt = 0; mt < 2; ++mt) {
            int row = mt * 16 + col;
            v2f a;
            a.x = sf[w][row][k0];
            a.y = sf[w][row][k0 + 1];
            #pragma unroll
            for (int nt = 0; nt < 4; ++nt)
                acc[mt][nt] = __builtin_amdgcn_wmma_f32_16x16x4_f32(
                    false, a, false, Bop[kk][nt], (short)0, acc[mt][nt],
                    false, false);
        }
    }

    // affine + relu + max over 32 points; D layout: VGPR r, lane L ->
    // (M = r + 8*(L>=16), N = L&15)  => shfl_xor(16) merges point halves.
    size_t plane = (size_t)YLc * XLc;
    size_t base  = (size_t)bb * COUTc * plane + (size_t)yi * XLc + (size_t)xi;
    #pragma unroll
    for (int nt = 0; nt < 4; ++nt) {
        float m = 0.f;  // relu lower bound
        #pragma unroll
        for (int mt = 0; mt < 2; ++mt)
            #pragma unroll
            for (int r = 0; r < 8; ++r)
                m = fmaxf(m, acc[mt][nt][r] * aco[nt] + bco[nt]);
        m = fmaxf(m, __shfl_xor(m, 16, 32));
        if (lane < 16) {
            int ch = nt * 16 + col;
            out[base + (size_t)ch * plane] = m;
        }
    }
}

extern "C" void kernel_launch(void* const* d_in, const int* in_sizes, int n_in,
                              void* d_out, int out_size, void* d_ws, size_t ws_size,
                              hipStream_t stream) {
    const float* pillars = (const float*)d_in[0];
    const int*   coors   = (const int*)d_in[1];
    const int*   npts    = (const int*)d_in[2];
    const float* W       = (const float*)d_in[3];
    const float* gamma   = (const float*)d_in[4];
    const float* beta    = (const float*)d_in[5];
    float* out = (float*)d_out;
    float* ws  = (float*)d_ws;   // [0..53] stats, [64..127] a, [128..191] b

    (void)in_sizes; (void)n_in; (void)ws_size;

    hipMemsetAsync(out, 0, (size_t)out_size * sizeof(float), stream);
    hipMemsetAsync(ws, 0, 64 * sizeof(float), stream);

    pillar_stats_kernel<<<5000, 256, 0, stream>>>(
        (const float4*)pillars, coors, npts, ws);
    pillar_finalize_kernel<<<1, 64, 0, stream>>>(
        ws, W, gamma, beta, ws + 64);
    pillar_main_kernel<<<10000, 128, 0, stream>>>(
        (const float4*)pillars, coors, npts, W, ws + 64, out);
}